// YOLO_63694364999712
// MI455X (gfx1250) — compile-verified
//
#include <hip/hip_runtime.h>
#include <hip/hip_bf16.h>

#define CONF_THR 0.25f
#define IOU_THR  0.45f
#define TOPK     1024
#define MAX_DET  300
#define MAX_WH   4096.0f
#define NBATCH   16
#define NANCH    25200
#define NCLS     80

typedef __attribute__((ext_vector_type(2))) float v2f;
typedef __attribute__((ext_vector_type(8))) float v8f;
typedef unsigned int u32x4 __attribute__((ext_vector_type(4)));
typedef int          i32x4 __attribute__((ext_vector_type(4)));
typedef int          i32x8 __attribute__((ext_vector_type(8)));

#if defined(__HIP_DEVICE_COMPILE__) && __has_builtin(__builtin_amdgcn_wmma_f32_16x16x4_f32)
#define HAVE_WMMA 1
#else
#define HAVE_WMMA 0
#endif

#if defined(__HIP_DEVICE_COMPILE__) && __has_builtin(__builtin_amdgcn_tensor_load_to_lds) && \
    __has_builtin(__builtin_amdgcn_s_wait_tensorcnt)
#define HAVE_TDM 1
#else
#define HAVE_TDM 0
#endif

// ---------------------------------------------------------------------------
// Stage 1: decode. The whole 256-anchor x 85-float block slab (87,040 B) is
// DMA'd into dynamic LDS by the Tensor Data Mover (one 1-D TENSOR_LOAD_TO_LDS
// per workgroup, tracked by TENSORcnt), then each lane reduces its anchor.
// Fallback (no TDM builtin): coalesced b128 global loads staged via LDS.
// ---------------------------------------------------------------------------
__global__ void decode_kernel(const float* __restrict__ pred,
                              float* __restrict__ boxes,
                              float* __restrict__ scores,
                              int* __restrict__ cls) {
  extern __shared__ float lds[];                 // 256 anchors * 85 floats
  const int tid  = threadIdx.x;
  const int lane = tid & 31;
  const int wave = tid >> 5;

  const long blockAnchor0 = (long)blockIdx.x * 256;
  const long anchor0      = blockAnchor0 + wave * 32;

#if HAVE_TDM
  if (wave == 0) {
    // ---- Tensor DMA descriptor (D#), per CDNA5 ISA ----
    // 1-D tile: 21760 fp32 elements (256 anchors x 85 ch) -> LDS offset 0.
    const unsigned elems = 256u * 85u;                      // 21760
    unsigned long long ga =
        (unsigned long long)(uintptr_t)(pred + blockAnchor0 * 85);
    u32x4 g0;
    g0[0] = 1u;                                             // count=1, user mode
    g0[1] = 0u;                                             // lds_addr (dyn LDS base)
    g0[2] = (unsigned)(ga & 0xFFFFFFFFu);                   // global_addr[31:0]
    g0[3] = (unsigned)((ga >> 32) & 0x01FFFFFFu)            // global_addr[56:32]
          | 0x80000000u;                                    // type=2 ("image")
    i32x8 g1;
    g1[0] = (int)(2u << 16);                                // data_size=4B, no mcast
    g1[1] = (int)((elems & 0xFFFFu) << 16);                 // tensor_dim0[15:0]
    g1[2] = (int)(((elems >> 16) & 0xFFFFu) | (1u << 16));  // dim0[31:16] | dim1 lo=1
    g1[3] = (int)(elems << 16);                             // dim1 hi=0 | tile_dim0
    g1[4] = 0;                                              // tile_dim1/2 unused
    g1[5] = (int)elems;                                     // dim0_stride[31:0]
    g1[6] = 0;                                              // stride hi / dim1_stride
    g1[7] = 0;
    i32x4 z4 = {0, 0, 0, 0};
#if __clang_major__ >= 23
    i32x8 z8 = {0, 0, 0, 0, 0, 0, 0, 0};
    __builtin_amdgcn_tensor_load_to_lds(g0, g1, z4, z4, z8, 0);
#else
    __builtin_amdgcn_tensor_load_to_lds(g0, g1, z4, z4, 0);
#endif
    __builtin_amdgcn_s_wait_tensorcnt(0);
  }
  __syncthreads();
#else
  {
    float* wls = lds + wave * (32 * 85);         // 2720 floats per wave
    const float4* __restrict__ src4 = (const float4*)(pred + anchor0 * 85);
    float4* w4 = (float4*)wls;
    for (int i = lane; i < 680; i += 32) {       // 680 float4 per wave
      __builtin_prefetch((const void*)(src4 + i + 680), 0, 0);
      w4[i] = src4[i];
    }
    __syncthreads();
  }
#endif

  const float* ap = lds + (wave * 32 + lane) * 85;
  float cx = ap[0], cy = ap[1], ww = ap[2], hh = ap[3], obj = ap[4];
  float best = ap[5];
  int bi = 0;
#pragma unroll 8
  for (int c = 1; c < NCLS; ++c) {
    float v = ap[5 + c];
    if (v > best) { best = v; bi = c; }
  }
  float s = obj * best;                 // argmax(obj*cls) == argmax(cls), obj>=0
  s = (s > CONF_THR) ? s : 0.0f;

  long g = anchor0 + lane;
  scores[g] = s;
  cls[g]    = bi;
  float hw = ww * 0.5f, hv = hh * 0.5f;
  boxes[g * 4 + 0] = cx - hw;
  boxes[g * 4 + 1] = cy - hv;
  boxes[g * 4 + 2] = cx + hw;
  boxes[g * 4 + 3] = cy + hv;
}

// ---------------------------------------------------------------------------
// Bitonic helpers (n = 1024, one element per thread, key desc + payload)
// ---------------------------------------------------------------------------
__device__ __forceinline__ void bitonic_step(float* key, int* pay, int tid,
                                             int j, int k) {
  __syncthreads();
  int ixj = tid ^ j;
  if (ixj > tid) {
    bool wantDesc = ((tid & k) == 0);            // k==0 -> descending everywhere
    float a = key[tid], b = key[ixj];
    bool sw = wantDesc ? (a < b) : (a > b);
    if (sw) {
      key[tid] = b; key[ixj] = a;
      int t = pay[tid]; pay[tid] = pay[ixj]; pay[ixj] = t;
    }
  }
}

__device__ void sort_desc_1024(float* key, int* pay, int tid) {
  for (int k = 2; k <= 1024; k <<= 1)
    for (int j = k >> 1; j > 0; j >>= 1)
      bitonic_step(key, pay, tid, j, k);
  __syncthreads();
}

__device__ void merge_desc_1024(float* key, int* pay, int tid) {
  for (int j = 512; j > 0; j >>= 1)
    bitonic_step(key, pay, tid, j, 0);
  __syncthreads();
}

// ---------------------------------------------------------------------------
// Stage 2: per-batch top-1024 (sorted desc) via streaming bitonic top-k.
// ---------------------------------------------------------------------------
__global__ void topk_kernel(const float* __restrict__ scores,
                            float* __restrict__ topS,
                            int* __restrict__ topI) {
  __shared__ float keyL[1024]; __shared__ int idxL[1024];
  __shared__ float keyT[1024]; __shared__ int idxT[1024];
  const int tid = threadIdx.x;
  const int b   = blockIdx.x;
  const float* sb = scores + (long)b * NANCH;

  keyL[tid] = sb[tid];            // tile 0 (tid < 1024 < 25200)
  idxL[tid] = tid;
  sort_desc_1024(keyL, idxL, tid);

  const int nTiles = (NANCH + 1023) / 1024;      // 25
  for (int t = 1; t < nTiles; ++t) {
    int g = t * 1024 + tid;
    keyT[tid] = (g < NANCH) ? sb[g] : -1.0f;     // pad below any real score
    idxT[tid] = (g < NANCH) ? g : 0;
    sort_desc_1024(keyT, idxT, tid);
    __syncthreads();
    // top-1024 of union: max(L[i], T[1023-i]) is bitonic
    float lk = keyL[tid];        int lp = idxL[tid];
    float tk = keyT[1023 - tid]; int tp = idxT[1023 - tid];
    float nk; int np;
    if (lk >= tk) { nk = lk; np = lp; } else { nk = tk; np = tp; }
    __syncthreads();
    keyL[tid] = nk; idxL[tid] = np;
    merge_desc_1024(keyL, idxL, tid);
  }
  topS[b * TOPK + tid] = keyL[tid];
  topI[b * TOPK + tid] = idxL[tid];
}

// ---------------------------------------------------------------------------
// Stage 3: NMS + final top-300. One 1024-thread block per batch.
// LDS: boxes/areas/scores (36 KB) + full 1024x1024 suppression bitmask
// (128 KB) -- possible thanks to CDNA5's 320 KB LDS per WGP.
// The denominator's area term is area[i] only on the WMMA path, per 16x16 tile with
// V_WMMA_F32_16X16X4_F32 (area_j meets a zero in A), one wave per tile.
// ---------------------------------------------------------------------------
__global__ void nms_kernel(const float* __restrict__ topS,
                           const int* __restrict__ topI,
                           const float* __restrict__ boxes,
                           const int* __restrict__ cls,
                           float* __restrict__ out) {
  extern __shared__ float smem[];
  float* x1s    = smem;
  float* y1s    = smem + 1024;
  float* x2s    = smem + 2048;
  float* y2s    = smem + 3072;
  float* areaS  = smem + 4096;
  float* scoreS = smem + 5120;
  int*   clsS   = (int*)(smem + 6144);
  float* key    = smem + 7168;
  int*   pay    = (int*)(smem + 8192);
  unsigned* mask      = (unsigned*)(smem + 9216);   // [1024][32] bit rows
  unsigned* keepWords = mask + 1024 * 32;           // [32]

  const int tid = threadIdx.x, b = blockIdx.x;
  const int lane = tid & 31, wave = tid >> 5;

  // ---- load candidates (class-offset boxes, like reference off_boxes) ----
  int gi = topI[b * TOPK + tid];
  gi = (gi < 0) ? 0 : ((gi >= NANCH) ? (NANCH - 1) : gi);
  float s = topS[b * TOPK + tid];
  long gg = (long)b * NANCH + gi;
  int c = cls[gg];
  float offv = (float)c * MAX_WH;
  float X1 = boxes[gg * 4 + 0] + offv;
  float Y1 = boxes[gg * 4 + 1] + offv;
  float X2 = boxes[gg * 4 + 2] + offv;
  float Y2 = boxes[gg * 4 + 3] + offv;
  x1s[tid] = X1; y1s[tid] = Y1; x2s[tid] = X2; y2s[tid] = Y2;
  areaS[tid]  = fmaxf(X2 - X1, 0.0f) * fmaxf(Y2 - Y1, 0.0f);
  scoreS[tid] = s;
  clsS[tid]   = c;
#pragma unroll
  for (int w = 0; w < 32; ++w) mask[tid * 32 + w] = 0u;
  __syncthreads();

  // ---- suppression matrix: 64x64 tiles of 16x16, one wave per tile ----
  for (int tile = wave; tile < 4096; tile += 32) {
    int I0 = (tile >> 6) << 4;
    int J0 = (tile & 63) << 4;
    int col = J0 + (lane & 15);
#if HAVE_WMMA
    // A (16x4): [area_i, 1, 0, 0]; B (4x16): [1; 0; area_j; 0]
    v2f aOp, bOp;
    aOp[0] = (lane < 16) ? areaS[I0 + lane] : 0.0f;
    aOp[1] = (lane < 16) ? 1.0f : 0.0f;
    bOp[0] = (lane < 16) ? 1.0f : areaS[J0 + (lane - 16)];
    bOp[1] = 0.0f;
    v8f acc = {};
    acc = __builtin_amdgcn_wmma_f32_16x16x4_f32(
        false, aOp, false, bOp, (short)0, acc, false, false);
#else
    float cArea = areaS[col];
#endif
    float cx1 = x1s[col], cy1 = y1s[col], cx2 = x2s[col], cy2 = y2s[col];
    int rbase = I0 + ((lane >> 4) << 3);   // matches C/D VGPR row mapping
#pragma unroll
    for (int r = 0; r < 8; ++r) {
      int row = rbase + r;
      float iw = fminf(x2s[row], cx2) - fmaxf(x1s[row], cx1);
      float ih = fminf(y2s[row], cy2) - fmaxf(y1s[row], cy1);
      iw = fmaxf(iw, 0.0f); ih = fmaxf(ih, 0.0f);
      float inter = iw * ih;
#if HAVE_WMMA
      float asum = acc[r];                 // area[row] only from WMMA
#else
      float asum = areaS[row] + cArea;
#endif
      float iou = inter / (asum - inter + 1e-7f);
      bool bit = (iou > IOU_THR) && (col > row);
      unsigned bal = (unsigned)__ballot(bit);   // wave32 ballot
      if (lane == 0)
        atomicOr(&mask[(I0 + r) * 32 + (J0 >> 5)],
                 (bal & 0xFFFFu) << (J0 & 16));
      if (lane == 16)
        atomicOr(&mask[(I0 + 8 + r) * 32 + (J0 >> 5)],
                 (bal >> 16) << (J0 & 16));
    }
  }
  __syncthreads();

  // ---- sequential suppression scan: wave 0, keep bits in registers ----
  if (tid < 32) {
    unsigned keepW = 0;
    for (int i2 = 0; i2 < 32; ++i2)
      keepW |= (scoreS[tid * 32 + i2] > CONF_THR ? 1u : 0u) << i2;
    for (int i = 0; i < 1024; ++i) {
      unsigned cur = __shfl(keepW, i >> 5, 32);
      if ((cur >> (i & 31)) & 1u)
        keepW &= ~mask[i * 32 + tid];
    }
    keepWords[tid] = keepW;
  }
  __syncthreads();

  // ---- final top-300 ----
  key[tid] = ((keepWords[tid >> 5] >> (tid & 31)) & 1u) ? scoreS[tid] : 0.0f;
  pay[tid] = tid;
  sort_desc_1024(key, pay, tid);
  if (tid < MAX_DET) {
    int j = pay[tid];
    float sc = key[tid];
    float ov = (float)clsS[j] * MAX_WH;
    float* o = out + ((long)b * MAX_DET + tid) * 6;
    o[0] = x1s[j] - ov;    // un-offset box (reference outputs cand_boxes)
    o[1] = y1s[j] - ov;
    o[2] = x2s[j] - ov;
    o[3] = y2s[j] - ov;
    o[4] = sc;
    o[5] = (float)clsS[j];
  }
}

// ---------------------------------------------------------------------------
extern "C" void kernel_launch(void* const* d_in, const int* in_sizes, int n_in,
                              void* d_out, int out_size, void* d_ws,
                              size_t ws_size, hipStream_t stream) {
  (void)in_sizes; (void)n_in; (void)out_size; (void)ws_size;
  const float* pred = (const float*)d_in[0];
  float* out = (float*)d_out;

  char* base = (char*)d_ws;
  size_t off = 0;
  auto carve = [&](size_t bytes) -> void* {
    void* p = base + off;
    off += (bytes + 255) & ~(size_t)255;
    return p;
  };
  const int nTot = NBATCH * NANCH;
  float* boxes  = (float*)carve((size_t)nTot * 4 * sizeof(float));
  float* scores = (float*)carve((size_t)nTot * sizeof(float));
  int*   clsArr = (int*)carve((size_t)nTot * sizeof(int));
  float* topS   = (float*)carve((size_t)NBATCH * TOPK * sizeof(float));
  int*   topI   = (int*)carve((size_t)NBATCH * TOPK * sizeof(int));

  size_t smem1 = (size_t)256 * 85 * sizeof(float);          // 87,040 B
  decode_kernel<<<nTot / 256, 256, smem1, stream>>>(pred, boxes, scores, clsArr);

  topk_kernel<<<NBATCH, 1024, 0, stream>>>(scores, topS, topI);

  size_t smem3 = (size_t)9 * 1024 * sizeof(float)           // boxes/scores/sort
               + (size_t)1024 * 32 * sizeof(unsigned)       // suppression mask
               + 32 * sizeof(unsigned);                     // keep words
  nms_kernel<<<NBATCH, 1024, smem3, stream>>>(topS, topI, boxes, clsArr, out);
}